// BoundaryLoss_71906342470021
// MI455X (gfx1250) — compile-verified
//
#include <hip/hip_runtime.h>

// Problem shape is fixed by the reference's setup_inputs().
#define BB 8
#define CC 20
#define HH 512
#define WW 512
#define TW 256                      // pixels (columns) per block tile
#define TP (TW + 4)                 // padded LDS target row (258 used)
#define NPART (BB * HH * (WW / TW)) // 8192 block partials
#define NPIX ((size_t)BB * HH * WW)

// ---- CDNA5 async global->LDS data mover -----------------------------------
// Per-lane fire-and-forget 16B copy: LDS dest offset in a VGPR, 64-bit global
// address in a VGPR pair, tracked by ASYNCcnt (CDNA5 ISA).
__device__ __forceinline__ void async_ld_b128(const float* g, float* lds_dst) {
  // addrspace(3)->flat cast puts the workgroup-relative LDS byte offset in the
  // low 32 bits (ISA 10.2: LDS aperture uses addr[31:0]).
  unsigned loff = (unsigned)(unsigned long long)lds_dst;
  asm volatile("global_load_async_to_lds_b128 %0, %1, off"
               :: "v"(loff), "v"(g)
               : "memory");
}

__device__ __forceinline__ void wait_async0() {
#if __has_builtin(__builtin_amdgcn_s_wait_asynccnt)
  __builtin_amdgcn_s_wait_asynccnt(0);
#else
  asm volatile("s_wait_asynccnt 0" ::: "memory");
#endif
}

// ---- wave32 XOR-butterfly reduction via ds_swizzle (no barriers, fixed order)
__device__ __forceinline__ float wave_reduce_add(float v) {
  // offset encoding (group-of-32): [14:10]=xor_mask, [9:5]=or_mask, [4:0]=and_mask
  v += __int_as_float(__builtin_amdgcn_ds_swizzle(__float_as_int(v), 0x401F)); // xor 16
  v += __int_as_float(__builtin_amdgcn_ds_swizzle(__float_as_int(v), 0x201F)); // xor 8
  v += __int_as_float(__builtin_amdgcn_ds_swizzle(__float_as_int(v), 0x101F)); // xor 4
  v += __int_as_float(__builtin_amdgcn_ds_swizzle(__float_as_int(v), 0x081F)); // xor 2
  v += __int_as_float(__builtin_amdgcn_ds_swizzle(__float_as_int(v), 0x041F)); // xor 1
  return v;
}

// ---- main kernel: one (b, h, 256-col tile) per block ----------------------
__global__ __launch_bounds__(256) void boundary_loss_main(
    const float* __restrict__ logits, const int* __restrict__ targets,
    float* __restrict__ partials) {
  __shared__ __align__(16) float sL[CC * TW]; // 20 KB logit tile
  __shared__ int   sT[3 * TP];                // 3 target rows w/ halo
  __shared__ float sR[8];                     // per-wave partials

  const int t     = threadIdx.x;
  const int wt    = blockIdx.x;   // 0..1
  const int h     = blockIdx.y;   // 0..511
  const int b     = blockIdx.z;   // 0..7
  const int wbase = wt * TW;

  // 1) Kick off async DMA of the 20 channel rows for this tile.
  //    20*256 floats = 1280 x 16B chunks = 5 per thread, full EXEC.
  const size_t chanStride = (size_t)HH * WW;
  const size_t base0 = (size_t)b * CC * chanStride + (size_t)h * WW + wbase;
#pragma unroll
  for (int k = 0; k < 5; ++k) {
    int i = t + k * 256;        // 0..1279
    int c = i >> 6;             // 64 chunks per channel row
    int j = (i & 63) << 2;      // float index within the 256-wide row
    async_ld_b128(logits + base0 + (size_t)c * chanStride + j, &sL[c * TW + j]);
  }

  // 2) Stage 3 clamped target rows (replicate padding) while DMA is in flight.
#pragma unroll
  for (int r = 0; r < 3; ++r) {
    int hr = min(max(h - 1 + r, 0), HH - 1);
    const int* trow = targets + ((size_t)b * HH + hr) * WW;
    for (int j = t; j < TW + 2; j += 256) {
      int col = min(max(wbase - 1 + j, 0), WW - 1);
      sT[r * TP + j] = trow[col];
    }
  }
  __syncthreads();

  // 3) Boundary weight: max over neighbor classes of integer Sobel magnitude.
  //    Center tap has zero weight in both kernels, so the 8 neighbors are the
  //    only candidate classes with a possible non-zero response. GVN collapses
  //    the 64 equality tests to the 28 unique pairs.
  int nb[8];
  nb[0] = sT[0 * TP + t];     nb[1] = sT[0 * TP + t + 1]; nb[2] = sT[0 * TP + t + 2];
  nb[3] = sT[1 * TP + t];     const int tc = sT[1 * TP + t + 1];
  nb[4] = sT[1 * TP + t + 2];
  nb[5] = sT[2 * TP + t];     nb[6] = sT[2 * TP + t + 1]; nb[7] = sT[2 * TP + t + 2];
  const int wxv[8] = {-1,  0,  1, -2, 2, -1, 0, 1};
  const int wyv[8] = {-1, -2, -1,  0, 0,  1, 2, 1};
  int best = 0;
#pragma unroll
  for (int k = 0; k < 8; ++k) {
    int c = nb[k];
    int gx = 0, gy = 0;
#pragma unroll
    for (int i = 0; i < 8; ++i) {
      int e = (nb[i] == c) ? 1 : 0;
      gx += wxv[i] * e;
      gy += wyv[i] * e;
    }
    int m2 = gx * gx + gy * gy;
    best = best > m2 ? best : m2;
  }
  float bmag = sqrtf((float)best);
  float wgt  = __expf(fminf(bmag * 3.0f, 5.0f)); // clip lower bound free (bmag>=0)

  // 4) Wait for the async logit tile, then stable logsumexp over C=20.
  wait_async0();
  __syncthreads();

  float l[CC];
#pragma unroll
  for (int c = 0; c < CC; ++c) l[c] = sL[c * TW + t]; // stride-1 -> conflict-free
  float m = l[0];
#pragma unroll
  for (int c = 1; c < CC; ++c) m = fmaxf(m, l[c]);
  float s = 0.0f;
#pragma unroll
  for (int c = 0; c < CC; ++c) s += __expf(l[c] - m);
  float lt  = sL[tc * TW + t];            // gather at target class (runtime idx)
  float nll = m + __logf(s) - lt;

  // 5) Deterministic reduction: wave32 butterfly, then fixed-order 8-way sum.
  float v = wave_reduce_add(wgt * nll);
  if ((t & 31) == 0) sR[t >> 5] = v;
  __syncthreads();
  if (t == 0) {
    float acc = 0.0f;
#pragma unroll
    for (int w = 0; w < 8; ++w) acc += sR[w];
    partials[((size_t)b * HH + h) * (WW / TW) + wt] = acc;
  }
}

// ---- finalize: fixed-order fp64 reduction of 8192 partials ----------------
__global__ __launch_bounds__(256) void boundary_loss_finalize(
    const float* __restrict__ partials, float* __restrict__ out) {
  __shared__ double sD[256];
  int t = threadIdx.x;
  double acc = 0.0;
  for (int i = t; i < NPART; i += 256) acc += (double)partials[i];
  sD[t] = acc;
  __syncthreads();
  for (int sft = 128; sft > 0; sft >>= 1) {
    if (t < sft) sD[t] += sD[t + sft];
    __syncthreads();
  }
  if (t == 0) out[0] = (float)(sD[0] / (double)NPIX);
}

extern "C" void kernel_launch(void* const* d_in, const int* in_sizes, int n_in,
                              void* d_out, int out_size, void* d_ws, size_t ws_size,
                              hipStream_t stream) {
  const float* logits  = (const float*)d_in[0];
  const int*   targets = (const int*)d_in[1];
  float* out      = (float*)d_out;
  float* partials = (float*)d_ws;   // needs 8192 * 4 = 32 KB

  dim3 grid(WW / TW, HH, BB);       // 2 x 512 x 8 = 8192 blocks
  boundary_loss_main<<<grid, 256, 0, stream>>>(logits, targets, partials);
  boundary_loss_finalize<<<1, 256, 0, stream>>>(partials, out);
}